// GAT_40372692582770
// MI455X (gfx1250) — compile-verified
//
#include <hip/hip_runtime.h>

#define D_FEAT 64
#define HC 128
#define NHEAD 2
#define CCH 64
#define NEG_SLOPE 0.2f

typedef __attribute__((ext_vector_type(2))) float v2f;
typedef __attribute__((ext_vector_type(8))) float v8f;

// ---------------------------------------------------------------- utilities

__global__ void fill_kernel(float* __restrict__ p, float v, int n) {
    int i = blockIdx.x * blockDim.x + threadIdx.x;
    if (i < n) p[i] = v;
}

__device__ __forceinline__ void atomicMaxF(float* addr, float val) {
    // float atomic-max via integer ordering trick (valid with -inf init)
    if (val >= 0.0f) atomicMax((int*)addr, __float_as_int(val));
    else             atomicMin((unsigned int*)addr, __float_as_uint(val));
}

// ------------------------------------------------- GEMM: Y = X@W + b (WMMA)
// X: [rows, 64] row-major, W: [64, 128] row-major, Y: [rows, 128]
// One wave32 computes one 16x16 tile of Y via V_WMMA_F32_16X16X4_F32.
// A (16x4 f32): lanes 0-15 rows, VGPR0={K0|K2}, VGPR1={K1|K3}
// B (4x16 f32): lanes 0-15 cols, same K split
// C/D (16x16 f32): VGPR j -> row j (lanes 0-15) / row j+8 (lanes 16-31)
__global__ void gemm64x128_wmma(const float* __restrict__ X,
                                const float* __restrict__ W,
                                const float* __restrict__ bvec,
                                float* __restrict__ Y, int rows) {
    const int wave = (blockIdx.x * blockDim.x + threadIdx.x) >> 5;
    const int lane = threadIdx.x & 31;
    const int tiles_m = rows >> 4;            // rows multiple of 16 (N=100000)
    const int tm = wave >> 3;                 // 128/16 = 8 tiles along N-dim
    const int tn = wave & 7;
    if (tm >= tiles_m) return;

    const int mrow  = (tm << 4) + (lane & 15);
    const int ncol  = (tn << 4) + (lane & 15);
    const int khalf = (lane >> 4) << 1;       // 0 or 2

    v8f acc = {};
#pragma unroll
    for (int k0 = 0; k0 < D_FEAT; k0 += 4) {
        const int kk = k0 + khalf;
        v2f a, b;
        const float* ap = X + (size_t)mrow * D_FEAT + kk;
        a.x = ap[0];
        a.y = ap[1];
        b.x = W[(size_t)(kk + 0) * HC + ncol];
        b.y = W[(size_t)(kk + 1) * HC + ncol];
        acc = __builtin_amdgcn_wmma_f32_16x16x4_f32(
            /*neg_a=*/false, a, /*neg_b=*/false, b,
            /*c_mod=*/(short)0, acc, /*reuse_a=*/false, /*reuse_b=*/false);
    }

    const float bb = bvec[ncol];
    const int rbase = (tm << 4) + ((lane >> 4) << 3);
#pragma unroll
    for (int j = 0; j < 8; ++j)
        Y[(size_t)(rbase + j) * HC + ncol] = acc[j] + bb;
}

// ------------------------------------- edge pass 1: scores + segment max
// one wave per edge; lane handles 4 channels (lanes 0-15 head0, 16-31 head1)
__global__ void score_kernel(const float* __restrict__ xl,
                             const float* __restrict__ xr,
                             const int* __restrict__ src,
                             const int* __restrict__ dst,
                             const float* __restrict__ att,   // [2,64] = 128
                             float* __restrict__ ebuf,        // [EE,2]
                             float* __restrict__ mmax,        // [N,2]
                             int E, int EE) {
    const int e    = (blockIdx.x * blockDim.x + threadIdx.x) >> 5;
    const int lane = threadIdx.x & 31;
    if (e >= EE) return;
    int s, d;
    if (e < E) { s = src[e]; d = dst[e]; } else { s = e - E; d = s; }

    const int c0 = lane << 2;
    const float4 zl = *(const float4*)(xl + (size_t)s * HC + c0);
    const float4 zr = *(const float4*)(xr + (size_t)d * HC + c0);
    const float4 at = *(const float4*)(att + c0);

    float p = 0.0f, z;
    z = zl.x + zr.x; z = (z > 0.0f) ? z : NEG_SLOPE * z; p += z * at.x;
    z = zl.y + zr.y; z = (z > 0.0f) ? z : NEG_SLOPE * z; p += z * at.y;
    z = zl.z + zr.z; z = (z > 0.0f) ? z : NEG_SLOPE * z; p += z * at.z;
    z = zl.w + zr.w; z = (z > 0.0f) ? z : NEG_SLOPE * z; p += z * at.w;

    // reduce within each 16-lane half (one head per half)
    p += __shfl_xor(p, 1);
    p += __shfl_xor(p, 2);
    p += __shfl_xor(p, 4);
    p += __shfl_xor(p, 8);

    if ((lane & 15) == 0) {
        const int h = lane >> 4;
        ebuf[(size_t)e * NHEAD + h] = p;
        atomicMaxF(&mmax[(size_t)d * NHEAD + h], p);
    }
}

// --------------------------- edge pass 2: a = exp(e - m[dst]), denom += a
__global__ void expsum_kernel(const int* __restrict__ src,
                              const int* __restrict__ dst,
                              float* __restrict__ ebuf,
                              const float* __restrict__ mmax,
                              float* __restrict__ denom,
                              int E, int EE) {
    int g = blockIdx.x * blockDim.x + threadIdx.x;
    if (g >= EE * NHEAD) return;
    const int e = g >> 1;
    const int h = g & 1;
    const int d = (e < E) ? dst[e] : (e - E);
    const float a = __expf(ebuf[g] - mmax[(size_t)d * NHEAD + h]);
    ebuf[g] = a;
    atomicAdd(&denom[(size_t)d * NHEAD + h], a);
}

// ------------------- edge pass 3: accum[dst] += (a/denom[dst]) * xl[src]
__global__ void scatter_kernel(const float* __restrict__ xl,
                               const float* __restrict__ ebuf,
                               const float* __restrict__ denom,
                               const int* __restrict__ src,
                               const int* __restrict__ dst,
                               float* __restrict__ accum,     // [N,128]
                               int E, int EE) {
    const int e    = (blockIdx.x * blockDim.x + threadIdx.x) >> 5;
    const int lane = threadIdx.x & 31;
    if (e >= EE) return;
    int s, d;
    if (e < E) { s = src[e]; d = dst[e]; } else { s = e - E; d = s; }

    const int h = lane >> 4;
    const float alpha = ebuf[(size_t)e * NHEAD + h] /
                        (denom[(size_t)d * NHEAD + h] + 1e-16f);
    const int c0 = lane << 2;
    const float4 v = *(const float4*)(xl + (size_t)s * HC + c0);
    float* o = accum + (size_t)d * HC + c0;
    atomicAdd(o + 0, alpha * v.x);
    atomicAdd(o + 1, alpha * v.y);
    atomicAdd(o + 2, alpha * v.z);
    atomicAdd(o + 3, alpha * v.w);
}

// -------------------------- head mean + bias (+ tanh-approx GELU between layers)
__global__ void finalize_kernel(const float* __restrict__ accum,
                                const float* __restrict__ bias,
                                float* __restrict__ out,
                                int apply_gelu, int total) {
    int g = blockIdx.x * blockDim.x + threadIdx.x;
    if (g >= total) return;
    const int node = g >> 6;
    const int c    = g & 63;
    float v = 0.5f * (accum[(size_t)node * HC + c] +
                      accum[(size_t)node * HC + CCH + c]) + bias[c];
    if (apply_gelu) {
        const float u = 0.7978845608028654f * (v + 0.044715f * v * v * v);
        v = 0.5f * v * (1.0f + tanhf(u));
    }
    out[g] = v;
}

// ---------------------------------------------------------------- launcher

extern "C" void kernel_launch(void* const* d_in, const int* in_sizes, int n_in,
                              void* d_out, int out_size, void* d_ws, size_t ws_size,
                              hipStream_t stream) {
    const float* x0   = (const float*)d_in[0];   // [N,64]
    const int*   conn = (const int*)  d_in[1];   // [2,E]
    const float* Wl   = (const float*)d_in[2];   // [L,64,128]
    const float* bl   = (const float*)d_in[3];   // [L,128]
    const float* Wr   = (const float*)d_in[4];   // [L,64,128]
    const float* br   = (const float*)d_in[5];   // [L,128]
    const float* att  = (const float*)d_in[6];   // [L,2,64]
    const float* bias = (const float*)d_in[7];   // [L,64]

    const int N  = in_sizes[0] / D_FEAT;
    const int E  = in_sizes[1] / 2;
    const int EE = E + N;                        // edges + self loops
    const int* src = conn;
    const int* dst = conn + E;

    // workspace carve-up (floats)
    float* ws    = (float*)d_ws;
    float* xl    = ws;                                   // [N,128]
    float* xr    = xl   + (size_t)N * HC;                // [N,128], reused as accum
    float* x1    = xr   + (size_t)N * HC;                // [N,64] inter-layer feats
    float* ebuf  = x1   + (size_t)N * D_FEAT;            // [EE,2]
    float* mmax  = ebuf + (size_t)EE * NHEAD;            // [N,2]
    float* denom = mmax + (size_t)N * NHEAD;             // [N,2]

    const dim3 blk(256);
    const int gemm_blocks = ((N / 16) * 8 + 7) / 8;                 // 8 waves/block
    const int edge_blocks = (int)(((long long)EE * 32 + 255) / 256); // wave/edge
    const int eh_blocks   = (EE * NHEAD + 255) / 256;
    const int nh          = N * NHEAD;
    const int nacc        = N * HC;
    const int fin_total   = N * D_FEAT;

    const int L = 2;
    const float* xin = x0;
    for (int l = 0; l < L; ++l) {
        const float* Wl_l = Wl + (size_t)l * D_FEAT * HC;
        const float* bl_l = bl + (size_t)l * HC;
        const float* Wr_l = Wr + (size_t)l * D_FEAT * HC;
        const float* br_l = br + (size_t)l * HC;
        const float* at_l = att + (size_t)l * NHEAD * CCH;
        const float* bi_l = bias + (size_t)l * CCH;

        gemm64x128_wmma<<<gemm_blocks, blk, 0, stream>>>(xin, Wl_l, bl_l, xl, N);
        gemm64x128_wmma<<<gemm_blocks, blk, 0, stream>>>(xin, Wr_l, br_l, xr, N);

        fill_kernel<<<(nh + 255) / 256, blk, 0, stream>>>(mmax, -__builtin_inff(), nh);
        fill_kernel<<<(nh + 255) / 256, blk, 0, stream>>>(denom, 0.0f, nh);

        score_kernel<<<edge_blocks, blk, 0, stream>>>(xl, xr, src, dst, at_l,
                                                      ebuf, mmax, E, EE);

        // xr dead after score pass -> reuse it as the output accumulator
        fill_kernel<<<(nacc + 255) / 256, blk, 0, stream>>>(xr, 0.0f, nacc);

        expsum_kernel<<<eh_blocks, blk, 0, stream>>>(src, dst, ebuf, mmax, denom, E, EE);

        scatter_kernel<<<edge_blocks, blk, 0, stream>>>(xl, ebuf, denom, src, dst,
                                                        xr, E, EE);

        float* outp = (l == L - 1) ? (float*)d_out : x1;
        finalize_kernel<<<(fin_total + 255) / 256, blk, 0, stream>>>(
            xr, bi_l, outp, (l < L - 1) ? 1 : 0, fin_total);

        xin = x1;
    }
}